// HyperGCN_18107582120687
// MI455X (gfx1250) — compile-verified
//
#include <hip/hip_runtime.h>
#include <hip/hip_bf16.h>

// ---------------------------------------------------------------------------
// HyperGCN 2-layer forward for MI455X (gfx1250, wave32, WMMA).
//   X1 = H @ W1                         (WMMA f32 16x16x4, HBM-bound)
//   S1 = spmm(A, X1)                    (edge-parallel f32 atomics, L2-resident)
//   X2 = relu(S1 + b1) @ W2             (WMMA f32, bias+relu fused into A-load)
//   S2 = spmm(A, X2)
//   out = log_softmax(relu(S2 + b2))    (one wave32 per row, shfl reductions)
// ---------------------------------------------------------------------------

typedef float v2f __attribute__((ext_vector_type(2)));
typedef float v8f __attribute__((ext_vector_type(8)));

#define GN   100000   // nodes
#define GD   512      // input features
#define GH1  16       // hidden
#define GC   40       // classes
#define GNNZ 3200000  // edges

// -------------------------------------------------------------------- zero
__global__ void hg_zero_kernel(float* __restrict__ p, int n) {
    int i = blockIdx.x * 256 + threadIdx.x;
    if (i < n) p[i] = 0.0f;
}

// -------------------------------------------------------------- GEMM1 (WMMA)
// X1[N,16] = H[N,512] @ W1[512,16].  One wave -> one 16-row tile.
// A layout (f32 16x4): lane L holds K = k0 + 2*(L>>4) + {0,1} of row (L&15).
// B layout (f32 4x16): VGPR v, lanes 0-15 K=v, lanes 16-31 K=v+2, N=L&15.
// C/D layout: VGPR v holds D[v + 8*(L>>4)][L&15].
__global__ void __launch_bounds__(256)
hg_gemm1_kernel(const float* __restrict__ H, const float* __restrict__ W1,
                float* __restrict__ X1) {
    __shared__ float ldsW[GD * GH1];          // 32 KB of the 320 KB/WGP LDS
    // cooperative stage of W1 into LDS (float4, fully coalesced)
    const float4* w4 = (const float4*)W1;
    float4*       l4 = (float4*)ldsW;
    for (int i = threadIdx.x; i < (GD * GH1) / 4; i += 256) l4[i] = w4[i];
    __syncthreads();

    const int wave = threadIdx.x >> 5;
    const int lane = threadIdx.x & 31;
    const int tile = blockIdx.x * 8 + wave;   // 16 rows per tile
    if (tile >= GN / 16) return;              // wave-uniform exit (EXEC stays full)

    const int m    = lane & 15;
    const int half = lane >> 4;
    const int row0 = tile * 16;

    const float* aptr = H + (long)(row0 + m) * GD + 2 * half;
    const int    bb   = (2 * half) * GH1 + m; // B column index == m

    v8f acc = {};
    #pragma unroll 4
    for (int k0 = 0; k0 < GD; k0 += 4) {
        v2f a = *(const v2f*)(aptr + k0);             // global_load_b64
        v2f b;
        b.x = ldsW[bb + k0 * GH1];                    // K = k0 + 2*half
        b.y = ldsW[bb + k0 * GH1 + GH1];              // K = k0 + 2*half + 1
        acc = __builtin_amdgcn_wmma_f32_16x16x4_f32(
                  false, a, false, b, (short)0, acc, false, false);
    }

    float* outp = X1 + (long)row0 * GH1 + m;
    #pragma unroll
    for (int v = 0; v < 8; ++v) outp[(v + 8 * half) * GH1] = acc[v];
}

// ------------------------------------------------------------------- SpMM1
// S1[r,f] += vals[e] * X1[c,f], 16 lanes per edge (64B coalesced gather).
__global__ void hg_spmm1_kernel(const int* __restrict__ rows,
                                const int* __restrict__ cols,
                                const float* __restrict__ vals,
                                const float* __restrict__ X1,
                                float* __restrict__ S1) {
    unsigned idx = blockIdx.x * 256u + threadIdx.x;
    unsigned e = idx >> 4;
    if (e >= GNNZ) return;
    int f = idx & 15;
    int c = cols[e], r = rows[e];
    float contrib = vals[e] * X1[(long)c * GH1 + f];
    __hip_atomic_fetch_add(&S1[(long)r * GH1 + f], contrib,
                           __ATOMIC_RELAXED, __HIP_MEMORY_SCOPE_AGENT);
}

// -------------------------------------------------------------- GEMM2 (WMMA)
// X2[N,40] = relu(S1 + b1) @ W2[16,40].  K=16 -> 4 WMMA steps, 3 column tiles.
__global__ void __launch_bounds__(256)
hg_gemm2_kernel(const float* __restrict__ S1, const float* __restrict__ b1,
                const float* __restrict__ W2, float* __restrict__ X2) {
    const int wave = threadIdx.x >> 5;
    const int lane = threadIdx.x & 31;
    const int tile = blockIdx.x * 8 + wave;
    if (tile >= GN / 16) return;

    const int n    = lane & 15;
    const int half = lane >> 4;
    const int row0 = tile * 16;

    v8f c0 = {}, c1 = {}, c2 = {};
    #pragma unroll
    for (int k0 = 0; k0 < GH1; k0 += 4) {
        const int kk = k0 + 2 * half;
        // fused bias + relu on the A operand
        v2f s = *(const v2f*)(S1 + (long)(row0 + n) * GH1 + kk);
        v2f a;
        a.x = fmaxf(s.x + b1[kk],     0.0f);
        a.y = fmaxf(s.y + b1[kk + 1], 0.0f);

        const float* wr0 = W2 + kk * GC;
        const float* wr1 = W2 + (kk + 1) * GC;
        v2f b0, b1v, b2v;
        b0.x  = wr0[n];       b0.y  = wr1[n];
        b1v.x = wr0[16 + n];  b1v.y = wr1[16 + n];
        b2v.x = (32 + n) < GC ? wr0[32 + n] : 0.0f;   // pad cols 40..47
        b2v.y = (32 + n) < GC ? wr1[32 + n] : 0.0f;

        c0 = __builtin_amdgcn_wmma_f32_16x16x4_f32(false, a, false, b0,
                                                   (short)0, c0, false, false);
        c1 = __builtin_amdgcn_wmma_f32_16x16x4_f32(false, a, false, b1v,
                                                   (short)0, c1, false, false);
        c2 = __builtin_amdgcn_wmma_f32_16x16x4_f32(false, a, false, b2v,
                                                   (short)0, c2, false, false);
    }

    #pragma unroll
    for (int v = 0; v < 8; ++v) {
        const long r = (long)(row0 + v + 8 * half) * GC;
        X2[r + n]      = c0[v];
        X2[r + 16 + n] = c1[v];
        if ((32 + n) < GC) X2[r + 32 + n] = c2[v];
    }
}

// ------------------------------------------------------------------- SpMM2
// S2[r,f] += vals[e] * X2[c,f], 40 lanes per edge (160B coalesced gather).
__global__ void hg_spmm2_kernel(const int* __restrict__ rows,
                                const int* __restrict__ cols,
                                const float* __restrict__ vals,
                                const float* __restrict__ X2,
                                float* __restrict__ S2) {
    unsigned idx = blockIdx.x * 256u + threadIdx.x;
    unsigned e = idx / 40u;                   // magic-multiply division
    if (e >= GNNZ) return;
    int f = (int)(idx - e * 40u);
    int c = cols[e], r = rows[e];
    float contrib = vals[e] * X2[(long)c * GC + f];
    __hip_atomic_fetch_add(&S2[(long)r * GC + f], contrib,
                           __ATOMIC_RELAXED, __HIP_MEMORY_SCOPE_AGENT);
}

// ------------------------------------------- fused relu + log_softmax (wave/row)
__global__ void hg_logsoftmax_kernel(const float* __restrict__ S2,
                                     const float* __restrict__ b2,
                                     float* __restrict__ out) {
    const int wave = threadIdx.x >> 5;
    const int lane = threadIdx.x & 31;
    const int row  = blockIdx.x * 8 + wave;
    if (row >= GN) return;

    const float* s = S2 + (long)row * GC;
    float v0 = fmaxf(s[lane] + b2[lane], 0.0f);                 // cols 0..31
    float v1 = (lane < 8) ? fmaxf(s[32 + lane] + b2[32 + lane], 0.0f)
                          : -INFINITY;                          // cols 32..39

    float m = fmaxf(v0, v1);
    #pragma unroll
    for (int off = 16; off > 0; off >>= 1)
        m = fmaxf(m, __shfl_xor(m, off, 32));

    float acc = expf(v0 - m) + ((lane < 8) ? expf(v1 - m) : 0.0f);
    #pragma unroll
    for (int off = 16; off > 0; off >>= 1)
        acc += __shfl_xor(acc, off, 32);

    float lse = logf(acc) + m;
    out[(long)row * GC + lane] = v0 - lse;
    if (lane < 8) out[(long)row * GC + 32 + lane] = v1 - lse;
}

// ---------------------------------------------------------------------------
extern "C" void kernel_launch(void* const* d_in, const int* in_sizes, int n_in,
                              void* d_out, int out_size, void* d_ws, size_t ws_size,
                              hipStream_t stream) {
    const float* H      = (const float*)d_in[0];
    const float* A_vals = (const float*)d_in[1];
    const float* W1     = (const float*)d_in[2];
    const float* b1     = (const float*)d_in[3];
    const float* W2     = (const float*)d_in[4];
    const float* b2     = (const float*)d_in[5];
    const int*   A_rows = (const int*)d_in[6];
    const int*   A_cols = (const int*)d_in[7];
    float*       out    = (float*)d_out;

    // workspace layout (floats): X1 | X2 | S1 | S2  (S1,S2 contiguous for zeroing)
    float* ws = (float*)d_ws;
    float* X1 = ws;                          // N*16
    float* X2 = ws + (long)GN * GH1;         // N*40
    float* S1 = ws + (long)GN * (GH1 + GC);  // N*16
    float* S2 = S1 + (long)GN * GH1;         // N*40

    const int zeroN = GN * (GH1 + GC);       // S1 and S2 together
    hg_zero_kernel<<<(zeroN + 255) / 256, 256, 0, stream>>>(S1, zeroN);

    const int tiles  = GN / 16;              // 6250
    const int gemmGrid = (tiles + 7) / 8;    // 8 waves (tiles) per 256-thread block
    hg_gemm1_kernel<<<gemmGrid, 256, 0, stream>>>(H, W1, X1);

    hg_spmm1_kernel<<<(GNNZ * 16 + 255) / 256, 256, 0, stream>>>(
        A_rows, A_cols, A_vals, X1, S1);

    hg_gemm2_kernel<<<gemmGrid, 256, 0, stream>>>(S1, b1, W2, X2);

    hg_spmm2_kernel<<<((long)GNNZ * 40 + 255) / 256, 256, 0, stream>>>(
        A_rows, A_cols, A_vals, X2, S2);

    hg_logsoftmax_kernel<<<(GN + 7) / 8, 256, 0, stream>>>(S2, b2, out);
}